// CDiceLoss_57784490001157
// MI455X (gfx1250) — compile-verified
//
#include <hip/hip_runtime.h>
#include <math.h>
#include <stdint.h>

typedef __attribute__((ext_vector_type(2))) float v2f;
typedef __attribute__((ext_vector_type(8))) float v8f;
typedef __attribute__((ext_vector_type(4))) float f4;
typedef __attribute__((ext_vector_type(4))) int   i4;

constexpr int  B = 4, C = 20, KNOWN = 16;
constexpr long K = 262144;            // 512*512
constexpr int  SPLIT = 32;            // K-splits per (b,c) in elementwise kernel
constexpr int  G_BLOCKS_PER_B = 64;   // Gram blocks per batch
constexpr int  KT = 512;              // K-chunk staged in LDS
constexpr int  LDS_STRIDE = KT + 4;   // +4 floats pad -> bank = (4*ch + k) % 64, conflict-free

// workspace layout (floats)
constexpr int OFF_XX  = 0;                    // B*C     sum x^2
constexpr int OFF_YY  = OFF_XX  + B * C;      // B*16    sum y^2
constexpr int OFF_XY  = OFF_YY  + B * KNOWN;  // B*16    sum x*y
constexpr int OFF_BCE = OFF_XY  + B * KNOWN;  // B*16    sum y*log(x)+(1-y)*log1p(-x)
constexpr int OFF_SX  = OFF_BCE + B * KNOWN;  // B*4     sum x (unknown channels)
constexpr int OFF_G   = OFF_SX  + B * (C - KNOWN); // B*32*32 padded Gram
constexpr int WS_FLOATS = OFF_G + B * 32 * 32;

#if __has_builtin(__builtin_amdgcn_global_load_async_to_lds_b128) && \
    __has_builtin(__builtin_amdgcn_s_wait_asynccnt)
#define USE_ASYNC_LDS 1
typedef __attribute__((address_space(1))) i4 gi4;   // global int4
typedef __attribute__((address_space(3))) i4 li4;   // LDS int4
#else
#define USE_ASYNC_LDS 0
#endif

__global__ void zero_ws_kernel(float* __restrict__ ws) {
    int i = blockIdx.x * blockDim.x + threadIdx.x;
    if (i < WS_FLOATS) ws[i] = 0.0f;
}

__device__ __forceinline__ float block_reduce256(float v, float* red) {
    int t = threadIdx.x;
    red[t] = v;
    __syncthreads();
    for (int off = 128; off > 0; off >>= 1) {
        if (t < off) red[t] += red[t + off];
        __syncthreads();
    }
    float r = red[0];
    __syncthreads();
    return r;
}

// ---------------- elementwise reductions (memory-bound streaming pass) ----------------
__global__ void __launch_bounds__(256) elemwise_kernel(const float* __restrict__ X,
                                                       const int* __restrict__ Y,
                                                       float* __restrict__ ws) {
    __shared__ float red[256];
    const int blk = blockIdx.x;
    const int s = blk % SPLIT;
    const int c = (blk / SPLIT) % C;
    const int b = blk / (SPLIT * C);
    const long kspan = K / SPLIT;                       // 8192 floats
    const long base = ((long)b * C + c) * K + (long)s * kspan;
    const f4* x4 = (const f4*)(X + base);
    const int kn = (int)(kspan / 4);                    // 2048 float4

    float sXX = 0.f, sYY = 0.f, sXY = 0.f, sBCE = 0.f, sX = 0.f;
    if (c < KNOWN) {
        const i4* y4 = (const i4*)(Y + base);
        for (int i = threadIdx.x; i < kn; i += 256) {
            f4 x = x4[i];
            i4 y = y4[i];
#pragma unroll
            for (int j = 0; j < 4; ++j) {
                float xv = x[j];
                float yv = (float)y[j];
                sXX += xv * xv;
                sYY += yv * yv;
                sXY += xv * yv;
                sBCE += yv * logf(xv) + (1.0f - yv) * log1pf(-xv);
            }
        }
    } else {
        for (int i = threadIdx.x; i < kn; i += 256) {
            f4 x = x4[i];
#pragma unroll
            for (int j = 0; j < 4; ++j) {
                float xv = x[j];
                sXX += xv * xv;
                sX += xv;
            }
        }
    }

    float rXX = block_reduce256(sXX, red);
    if (c < KNOWN) {
        float rYY = block_reduce256(sYY, red);
        float rXY = block_reduce256(sXY, red);
        float rBCE = block_reduce256(sBCE, red);
        if (threadIdx.x == 0) {
            atomicAdd(&ws[OFF_XX + b * C + c], rXX);
            atomicAdd(&ws[OFF_YY + b * KNOWN + c], rYY);
            atomicAdd(&ws[OFF_XY + b * KNOWN + c], rXY);
            atomicAdd(&ws[OFF_BCE + b * KNOWN + c], rBCE);
        }
    } else {
        float rX = block_reduce256(sX, red);
        if (threadIdx.x == 0) {
            atomicAdd(&ws[OFF_XX + b * C + c], rXX);
            atomicAdd(&ws[OFF_SX + b * (C - KNOWN) + (c - KNOWN)], rX);
        }
    }
}

// ---------------- Gram matrix via V_WMMA_F32_16X16X4_F32, async-LDS double-buffered ----
// Symmetric Gram: the ISA A and B lane layouts for 16x4 f32 coincide, so one v2f
// per lane feeds both operands. 20 channels padded to 32 -> tiles (0,0),(0,1),(1,1).

// stage one 20 x 512 chunk into an LDS buffer (256 threads, 10 float4 each)
__device__ __forceinline__ void stage_chunk(const float* __restrict__ Xb, long kb,
                                            float* __restrict__ buf, int tid) {
    for (int idx = tid; idx < C * (KT / 4); idx += 256) {
        int row = idx >> 7;            // idx / 128
        int col4 = idx & 127;
        const float* gsrc = Xb + (long)row * K + kb + (long)col4 * 4;
        float* ldst = &buf[row * LDS_STRIDE + col4 * 4];
#if USE_ASYNC_LDS
        __builtin_amdgcn_global_load_async_to_lds_b128(
            (gi4*)(uintptr_t)gsrc,
            (li4*)(uint32_t)(uintptr_t)ldst,   // low 32 bits of generic LDS ptr = LDS offset
            0, 0);
#else
        *(f4*)ldst = *(const f4*)gsrc;
#endif
    }
}

__global__ void __launch_bounds__(256) gram_kernel(const float* __restrict__ X,
                                                   float* __restrict__ ws) {
    __shared__ float lds[2 * C * LDS_STRIDE];   // 2 * 20 * 516 * 4B = 82560 B

    const int blk = blockIdx.x;
    const int b = blk / G_BLOCKS_PER_B;
    const int s = blk % G_BLOCKS_PER_B;
    const long kspan = K / G_BLOCKS_PER_B;      // 4096
    const long kbeg = (long)s * kspan;
    const float* Xb = X + (long)b * C * K;
    constexpr int NCHUNK = (int)(4096 / KT);    // 8

    const int tid = threadIdx.x;
    const int lane = tid & 31;
    const int wave = tid >> 5;
    const int m = lane & 15;      // N/col (and A-row channel)
    const int hk = lane >> 4;     // half-K selector
    const int ch1 = 16 + m;
    const float msk1 = (ch1 < C) ? 1.0f : 0.0f;
    const int ch1c = (ch1 < C) ? ch1 : (C - 1);

    v8f acc00 = {};   // G[0..15][0..15]
    v8f acc01 = {};   // G[0..15][16..31]
    v8f acc11 = {};   // G[16..31][16..31]

    // prologue: stage chunk 0 into buffer 0
    stage_chunk(Xb, kbeg, &lds[0], tid);

    for (int i = 0; i < NCHUNK; ++i) {
#if USE_ASYNC_LDS
        __builtin_amdgcn_s_wait_asynccnt(0);    // this wave's DMA into buf[i&1] done
#endif
        __syncthreads();                        // whole block's chunk i resident

        // kick off DMA for chunk i+1 into the other buffer; that buffer was last
        // read in iteration i-1, whose compute finished before this barrier.
        if (i + 1 < NCHUNK)
            stage_chunk(Xb, kbeg + (long)(i + 1) * KT, &lds[((i + 1) & 1) * C * LDS_STRIDE], tid);

        const float* buf = &lds[(i & 1) * C * LDS_STRIDE];
        // each wave: 16 k-steps of 4 -> 64 k-values; 8 waves cover 512
#pragma unroll 4
        for (int t = 0; t < 16; ++t) {
            int k0 = (wave * 16 + t) * 4 + 2 * hk;
            v2f a0 = *(const v2f*)(&buf[m * LDS_STRIDE + k0]);
            v2f a1 = *(const v2f*)(&buf[ch1c * LDS_STRIDE + k0]);
            a1 *= msk1;   // zero-pad channels 20..31
            acc00 = __builtin_amdgcn_wmma_f32_16x16x4_f32(false, a0, false, a0,
                                                          (short)0, acc00, false, false);
            acc01 = __builtin_amdgcn_wmma_f32_16x16x4_f32(false, a0, false, a1,
                                                          (short)0, acc01, false, false);
            acc11 = __builtin_amdgcn_wmma_f32_16x16x4_f32(false, a1, false, a1,
                                                          (short)0, acc11, false, false);
        }
    }

    // C/D layout: col = lane&15, row = vgpr + 8*(lane>>4)
    float* G = ws + OFF_G + b * 1024;
#pragma unroll
    for (int v = 0; v < 8; ++v) {
        int row = v + 8 * hk;
        atomicAdd(&G[row * 32 + m], acc00[v]);
        atomicAdd(&G[row * 32 + 16 + m], acc01[v]);
        atomicAdd(&G[(16 + row) * 32 + 16 + m], acc11[v]);
    }
}

// ---------------- finalize: combine scalars ----------------
__global__ void __launch_bounds__(256) finalize_kernel(const float* __restrict__ ws,
                                                       float* __restrict__ out) {
    __shared__ float red[256];
    const int tid = threadIdx.x;
    const float invBK = 1.0f / ((float)B * (float)K);

    float p1 = 0.f, p2 = 0.f, p3 = 0.f;

    if (tid < KNOWN) {                       // dice + bce per known channel
        int c = tid;
        float dice = 0.f, bces = 0.f;
        for (int b = 0; b < B; ++b) {
            float xx = ws[OFF_XX + b * C + c];
            float yy = ws[OFF_YY + b * KNOWN + c];
            float xy = ws[OFF_XY + b * KNOWN + c];
            dice += 1.0f - (xy + 1.0f) / (xx + yy + 1.0f);
            bces += ws[OFF_BCE + b * KNOWN + c];
        }
        p1 = dice / (float)B - bces * invBK;
    } else if (tid < C) {                    // unknown-channel penalty
        int ci = tid - KNOWN;
        float sx = 0.f;
        for (int b = 0; b < B; ++b) sx += ws[OFF_SX + b * (C - KNOWN) + ci];
        float mval = sx * invBK;
        float t = fminf(fmaxf(mval * 50.0f, 0.0f), 1.0f);
        p2 = -logf(t);
    }

    for (int idx = tid; idx < C * C; idx += 256) {   // off-diagonal ratio sum
        int i = idx / C, j = idx % C;
        if (i != j) {
            float r = 0.f;
            for (int b = 0; b < B; ++b) {
                int ii = i, jj = j;
                if (ii >= 16 && jj < 16) { int t2 = ii; ii = jj; jj = t2; }  // symmetry
                float g = ws[OFF_G + b * 1024 + ii * 32 + jj];
                float si = ws[OFF_XX + b * C + i];
                float sj = ws[OFF_XX + b * C + j];
                r += (g + 1.0f) / (si + sj + 1.0f);
            }
            p3 += r / (float)B;
        }
    }

    float r1 = block_reduce256(p1, red);
    float r2 = block_reduce256(p2, red);
    float r3 = block_reduce256(p3, red);

    if (tid == 0) {
        float loss1 = r1 / (float)KNOWN;
        float loss2 = r2 / (float)(C - KNOWN);
        float loss3 = r3 / (float)(C * (C - 1));
        float loss = 0.1f * (loss1 + loss2 + loss3);
        out[0] = loss;
        out[1] = loss1;
        out[2] = loss2;
        out[3] = loss3;
    }
}

extern "C" void kernel_launch(void* const* d_in, const int* in_sizes, int n_in,
                              void* d_out, int out_size, void* d_ws, size_t ws_size,
                              hipStream_t stream) {
    const float* X = (const float*)d_in[0];   // logit  (4,20,512,512) f32
    const int*   Y = (const int*)d_in[1];     // label  (4,20,512,512) i32
    // d_in[2] (class_lst) unused by the reference
    float* ws = (float*)d_ws;
    float* out = (float*)d_out;

    zero_ws_kernel<<<(WS_FLOATS + 255) / 256, 256, 0, stream>>>(ws);
    elemwise_kernel<<<B * C * SPLIT, 256, 0, stream>>>(X, Y, ws);
    gram_kernel<<<B * G_BLOCKS_PER_B, 256, 0, stream>>>(X, ws);
    finalize_kernel<<<1, 256, 0, stream>>>(ws, out);
}